// HybridLoss_85152021611307
// MI455X (gfx1250) — compile-verified
//
#include <hip/hip_runtime.h>
#include <hip/hip_bf16.h>
#include <math.h>

typedef __attribute__((ext_vector_type(2))) float v2f;
typedef __attribute__((ext_vector_type(8))) float v8f;

#define BOX_THREADS 32

// Build the 8 corners of an oriented box from 10 params:
// center(3), dims(3), quat(w,x,y,z). Matches reference _SIGNS ordering:
// k = 4*ix + 2*iy + iz, bit set => +1, clear => -1.
__device__ __forceinline__ void make_corners(const float p[10], float c[8][3]) {
    float cx = p[0], cy = p[1], cz = p[2];
    float hx = 0.5f * p[3], hy = 0.5f * p[4], hz = 0.5f * p[5];
    float w = p[6], x = p[7], y = p[8], z = p[9];
    float inv = 1.0f / (sqrtf(w * w + x * x + y * y + z * z) + 1e-6f);
    w *= inv; x *= inv; y *= inv; z *= inv;
    float R00 = 1.0f - 2.0f * (y * y + z * z);
    float R01 = 2.0f * (x * y - w * z);
    float R02 = 2.0f * (x * z + w * y);
    float R10 = 2.0f * (x * y + w * z);
    float R11 = 1.0f - 2.0f * (x * x + z * z);
    float R12 = 2.0f * (y * z - w * x);
    float R20 = 2.0f * (x * z - w * y);
    float R21 = 2.0f * (y * z + w * x);
    float R22 = 1.0f - 2.0f * (x * x + y * y);
#pragma unroll
    for (int k = 0; k < 8; ++k) {
        float ox = ((k >> 2) & 1) ? hx : -hx;
        float oy = ((k >> 1) & 1) ? hy : -hy;
        float oz = (k & 1) ? hz : -hz;
        c[k][0] = R00 * ox + R01 * oy + R02 * oz + cx;
        c[k][1] = R10 * ox + R11 * oy + R12 * oz + cy;
        c[k][2] = R20 * ox + R21 * oy + R22 * oz + cz;
    }
}

// One thread per box. Exact min-cost assignment on the 8x8 L1 cost matrix
// via bitmask DP over 2^8 subsets (replaces the reference's brute-force
// 40320-permutation matmul; the output only needs the min assignment cost).
// LDS tables are mask-major (stride 32) so the uniform-mask DP loop hits 32
// consecutive banks (conflict-free across the 64 LDS banks).
__global__ void box_loss_kernel(const float* __restrict__ pred,
                                const float* __restrict__ gt,
                                float* __restrict__ losses, int B) {
    __shared__ float cost_lds[64 * BOX_THREADS];  // [i*8+j][lane]
    __shared__ float f_lds[256 * BOX_THREADS];    // [mask][lane]

    int tid = threadIdx.x;
    int b = blockIdx.x * BOX_THREADS + tid;
    if (b >= B) return;

    float pp[10], gp[10];
#pragma unroll
    for (int i = 0; i < 10; ++i) {
        pp[i] = pred[b * 10 + i];
        gp[i] = gt[b * 10 + i];
    }

    float pc[8][3], gc[8][3];
    make_corners(pp, pc);
    make_corners(gp, gc);

    float diag = sqrtf(gp[3] * gp[3] + gp[4] * gp[4] + gp[5] * gp[5]);

    // 8x8 L1 cost matrix -> LDS
#pragma unroll
    for (int i = 0; i < 8; ++i) {
#pragma unroll
        for (int j = 0; j < 8; ++j) {
            float d = fabsf(pc[i][0] - gc[j][0]) +
                      fabsf(pc[i][1] - gc[j][1]) +
                      fabsf(pc[i][2] - gc[j][2]);
            cost_lds[(i * 8 + j) * BOX_THREADS + tid] = d;
        }
    }

    // Bitmask DP: f[mask] = cost of assigning pred rows 0..popcnt-1 to the
    // gt-corner set `mask`. Masks processed in increasing order.
    f_lds[0 * BOX_THREADS + tid] = 0.0f;
    for (int mask = 1; mask < 256; ++mask) {
        int i = __popc(mask) - 1;  // uniform across lanes (SALU)
        float best = INFINITY;
#pragma unroll
        for (int j = 0; j < 8; ++j) {
            if (mask & (1 << j)) {
                float v = f_lds[(mask ^ (1 << j)) * BOX_THREADS + tid] +
                          cost_lds[(i * 8 + j) * BOX_THREADS + tid];
                best = fminf(best, v);
            }
        }
        f_lds[mask * BOX_THREADS + tid] = best;
    }

    float min_assign = f_lds[255 * BOX_THREADS + tid];
    losses[b] = min_assign / (8.0f * (diag + 1e-6f));
}

// Single-wave mean reduction using V_WMMA_F32_16X16X4_F32.
// A = 16x4 ones, B = a 4x16 chunk of 64 values, accumulated into one f32
// 16x16 C. Every row of D is the running per-column sum, so the grand total
// equals the sum of row 0 (C VGPR0, lanes 0-15 per the ISA layout). The
// total is invariant to how the 64 chunk values are packed into B (each
// appears exactly once), so we pack them as one contiguous float2 per lane
// -> a single unconditional global_load_b64 per chunk, no exec masking.
__global__ void reduce_mean_wmma(const float* __restrict__ vals,
                                 float* __restrict__ out, int n) {
    int lane = threadIdx.x;  // 32 lanes, 1 wave
    int nfull = n & ~63;     // multiple-of-64 prefix handled by WMMA

    v8f c = {};
    v2f a;
    a.x = 1.0f;
    a.y = 1.0f;
    const v2f* vals2 = (const v2f*)vals;
    for (int base = 0; base < nfull; base += 64) {
        v2f bvec = vals2[(base >> 1) + lane];  // 256B-contiguous per chunk
        // 8 args: (neg_a, A, neg_b, B, c_mod, C, reuse_a, reuse_b)
        c = __builtin_amdgcn_wmma_f32_16x16x4_f32(
            false, a, false, bvec, (short)0, c, false, false);
    }

    // Scalar tail (empty when n % 64 == 0), reduced across all 32 lanes.
    float extra = 0.0f;
    for (int i = nfull + lane; i < n; i += 32) extra += vals[i];
#pragma unroll
    for (int off = 16; off >= 1; off >>= 1) extra += __shfl_xor(extra, off, 32);

    // Sum row 0 of the accumulator across its 16 lanes.
    float r = c[0];
#pragma unroll
    for (int off = 8; off >= 1; off >>= 1) r += __shfl_xor(r, off, 32);

    if (lane == 0) out[0] = (r + extra) / (float)n;
}

extern "C" void kernel_launch(void* const* d_in, const int* in_sizes, int n_in,
                              void* d_out, int out_size, void* d_ws, size_t ws_size,
                              hipStream_t stream) {
    (void)n_in;
    (void)out_size;
    (void)ws_size;
    const float* pred = (const float*)d_in[0];
    const float* gt = (const float*)d_in[1];
    int B = in_sizes[0] / 10;

    float* ws = (float*)d_ws;  // B per-box losses

    int blocks = (B + BOX_THREADS - 1) / BOX_THREADS;
    box_loss_kernel<<<blocks, BOX_THREADS, 0, stream>>>(pred, gt, ws, B);
    reduce_mean_wmma<<<1, 32, 0, stream>>>(ws, (float*)d_out, B);
}